// SVDO_84628035600386
// MI455X (gfx1250) — compile-verified
//
#include <hip/hip_runtime.h>
#include <hip/hip_bf16.h>
#include <math.h>

// ---------------------------------------------------------------------------
// SVDO: M = x @ W^T + b  (B x 9, viewed as B x 3 x 3), then nearest rotation
// R = U diag(1,1,det(U V^T)) V^T from the 3x3 SVD, fused in one kernel.
//
// GEMM side: memory-bound (512 MB of x). One wave owns a 16-row tile and runs
// V_WMMA_F32_16X16X4_F32 over K=512 (fp32-exact). B-operand fragments of the
// tiny W (9x512) are pre-staged in LDS in per-lane operand order.
// ---------------------------------------------------------------------------

typedef float v2f __attribute__((ext_vector_type(2)));
typedef float v4f __attribute__((ext_vector_type(4)));
typedef float v8f __attribute__((ext_vector_type(8)));

#define K_DIM 512
#define TILE_M 16
#define WAVES_PER_BLOCK 8
#define THREADS (WAVES_PER_BLOCK * 32)
#define NGROUPS (K_DIM / 8)   // 64 groups of 8 K-columns -> 2 WMMAs each

// ---- compile-time-indexed 3x3 Jacobi rotation on S (symmetric), V <- V*J ----
template <int p, int q>
__device__ __forceinline__ void jacobi_rot(float (&S)[3][3], float (&V)[3][3]) {
  float apq = S[p][q];
  if (fabsf(apq) > 1e-30f) {
    float tau = (S[q][q] - S[p][p]) / (2.0f * apq);
    float t = copysignf(1.0f, tau) / (fabsf(tau) + sqrtf(1.0f + tau * tau));
    float c = rsqrtf(1.0f + t * t);
    float s = t * c;
#pragma unroll
    for (int k = 0; k < 3; ++k) {        // S <- S * J
      float skp = S[k][p], skq = S[k][q];
      S[k][p] = c * skp - s * skq;
      S[k][q] = s * skp + c * skq;
    }
#pragma unroll
    for (int k = 0; k < 3; ++k) {        // S <- J^T * S
      float spk = S[p][k], sqk = S[q][k];
      S[p][k] = c * spk - s * sqk;
      S[q][k] = s * spk + c * sqk;
    }
#pragma unroll
    for (int k = 0; k < 3; ++k) {        // V <- V * J
      float vkp = V[k][p], vkq = V[k][q];
      V[k][p] = c * vkp - s * vkq;
      V[k][q] = s * vkp + c * vkq;
    }
  }
}

template <int a, int b>
__device__ __forceinline__ void sort2(float (&lam)[3], float (&V)[3][3]) {
  if (lam[a] < lam[b]) {
    float t = lam[a]; lam[a] = lam[b]; lam[b] = t;
#pragma unroll
    for (int k = 0; k < 3; ++k) {
      float tv = V[k][a]; V[k][a] = V[k][b]; V[k][b] = tv;
    }
  }
}

__global__ __launch_bounds__(THREADS) void svdo_fused_kernel(
    const float* __restrict__ x, const float* __restrict__ W,
    const float* __restrict__ bias, float* __restrict__ out, int n_tiles) {
  // Per-lane B-operand fragments for all 64 K-groups x 2 fragments: 32 KB.
  __shared__ v2f bfrag[NGROUPS * 2 * 32];
  // Per-wave 16x16 C tile for the lane-transpose into the SVD stage.
  __shared__ float ctile[WAVES_PER_BLOCK][TILE_M][18];  // pad 18: bank-clean

  const int tid  = threadIdx.x;
  const int lane = tid & 31;
  const int wave = tid >> 5;
  const int nlo  = lane & 15;    // output column N (9 real, 16 padded)
  const int half = lane >> 4;    // which half-wave (K-slot pair)

  // ---- Stage W into LDS in exact WMMA-B per-lane operand order -------------
  // K-slot mapping for group j, fragment f:
  //   lanes 0-15  : cols (8j + 2f,     8j + 2f + 1)
  //   lanes 16-31 : cols (8j + 4 + 2f, 8j + 4 + 2f + 1)
  for (int idx = tid; idx < NGROUPS * 2 * 32; idx += THREADS) {
    int j = idx >> 6;
    int f = (idx >> 5) & 1;
    int l = idx & 31;
    int n = l & 15;
    int kb = 8 * j + ((l & 16) ? 4 : 0) + 2 * f;
    v2f v = {0.f, 0.f};
    if (n < 9) {
      v.x = W[n * K_DIM + kb];
      v.y = W[n * K_DIM + kb + 1];
    }
    bfrag[idx] = v;
  }
  float bv = (nlo < 9) ? bias[nlo] : 0.f;
  __syncthreads();

  for (int tile = blockIdx.x * WAVES_PER_BLOCK + wave; tile < n_tiles;
       tile += gridDim.x * WAVES_PER_BLOCK) {
    const int row0 = tile * TILE_M;

    // Lane r streams row (row0+r) cols 8j..8j+3; lane 16+r cols 8j+4..8j+7.
    const v4f* __restrict__ xp = reinterpret_cast<const v4f*>(
        x + (size_t)(row0 + nlo) * K_DIM + (half << 2));

    v8f c0 = {0.f, 0.f, 0.f, 0.f, 0.f, 0.f, 0.f, 0.f};
    v8f c1 = {0.f, 0.f, 0.f, 0.f, 0.f, 0.f, 0.f, 0.f};
#pragma unroll 4
    for (int j = 0; j < NGROUPS; ++j) {
      v4f X = __builtin_nontemporal_load(xp + 2 * j);  // global_load_b128 (NT)
      v2f b0 = bfrag[(2 * j + 0) * 32 + lane];
      v2f b1 = bfrag[(2 * j + 1) * 32 + lane];
      v2f a0 = {X.x, X.y};
      v2f a1 = {X.z, X.w};
      // D = A(16x4 f32) * B(4x16 f32) + C, two independent accumulation chains
      c0 = __builtin_amdgcn_wmma_f32_16x16x4_f32(false, a0, false, b0,
                                                 (short)0, c0, false, false);
      c1 = __builtin_amdgcn_wmma_f32_16x16x4_f32(false, a1, false, b1,
                                                 (short)0, c1, false, false);
    }

    __syncthreads();  // protect previous iteration's ctile reads (WAR)

    // C/D layout: VGPR j, lanes 0-15 -> (M=j, N=lane); lanes 16-31 -> (M=j+8)
#pragma unroll
    for (int j = 0; j < 8; ++j) {
      float cv = c0[j] + c1[j] + bv;
      ctile[wave][j + (half << 3)][nlo] = cv;
    }

    __syncthreads();  // make ctile writes visible across lanes

    if (lane < 16) {
      // ---- gather this lane's 3x3 M --------------------------------------
      float M3[3][3];
#pragma unroll
      for (int i = 0; i < 3; ++i)
#pragma unroll
        for (int j = 0; j < 3; ++j) M3[i][j] = ctile[wave][lane][i * 3 + j];

      // ---- S = M^T M, Jacobi eigendecomposition S = V diag(lam) V^T ------
      float S[3][3], V[3][3];
#pragma unroll
      for (int i = 0; i < 3; ++i)
#pragma unroll
        for (int j = 0; j < 3; ++j) {
          S[i][j] = M3[0][i] * M3[0][j] + M3[1][i] * M3[1][j] +
                    M3[2][i] * M3[2][j];
          V[i][j] = (i == j) ? 1.f : 0.f;
        }
#pragma unroll
      for (int sweep = 0; sweep < 4; ++sweep) {
        jacobi_rot<0, 1>(S, V);
        jacobi_rot<0, 2>(S, V);
        jacobi_rot<1, 2>(S, V);
      }
      float lam[3] = {S[0][0], S[1][1], S[2][2]};
      sort2<0, 1>(lam, V);
      sort2<0, 2>(lam, V);
      sort2<1, 2>(lam, V);   // lam descending, V columns follow

      // ---- U from B = M*V (columns = sigma_i * u_i) ----------------------
      float B0[3], B1[3];
#pragma unroll
      for (int i = 0; i < 3; ++i) {
        B0[i] = M3[i][0] * V[0][0] + M3[i][1] * V[1][0] + M3[i][2] * V[2][0];
        B1[i] = M3[i][0] * V[0][1] + M3[i][1] * V[1][1] + M3[i][2] * V[2][1];
      }
      float inv0 = rsqrtf(B0[0] * B0[0] + B0[1] * B0[1] + B0[2] * B0[2] + 1e-30f);
      float u0[3] = {B0[0] * inv0, B0[1] * inv0, B0[2] * inv0};
      float d01 = u0[0] * B1[0] + u0[1] * B1[1] + u0[2] * B1[2];
      float w1[3] = {B1[0] - d01 * u0[0], B1[1] - d01 * u0[1], B1[2] - d01 * u0[2]};
      float inv1 = rsqrtf(w1[0] * w1[0] + w1[1] * w1[1] + w1[2] * w1[2] + 1e-30f);
      float u1[3] = {w1[0] * inv1, w1[1] * inv1, w1[2] * inv1};
      float u2[3] = {u0[1] * u1[2] - u0[2] * u1[1],
                     u0[2] * u1[0] - u0[0] * u1[2],
                     u0[0] * u1[1] - u0[1] * u1[0]};
      // det(U') = +1 by construction; reference flip becomes s = sign(det V).
      float detV = V[0][0] * (V[1][1] * V[2][2] - V[1][2] * V[2][1]) -
                   V[0][1] * (V[1][0] * V[2][2] - V[1][2] * V[2][0]) +
                   V[0][2] * (V[1][0] * V[2][1] - V[1][1] * V[2][0]);
      float sgn = (detV < 0.f) ? -1.f : 1.f;

      // ---- R = U diag(1,1,sgn) V^T ---------------------------------------
      size_t base = (size_t)(row0 + lane) * 9;
#pragma unroll
      for (int i = 0; i < 3; ++i)
#pragma unroll
        for (int j = 0; j < 3; ++j) {
          float r = u0[i] * V[j][0] + u1[i] * V[j][1] + sgn * u2[i] * V[j][2];
          out[base + i * 3 + j] = r;
        }
    }
  }
}

extern "C" void kernel_launch(void* const* d_in, const int* in_sizes, int n_in,
                              void* d_out, int out_size, void* d_ws,
                              size_t ws_size, hipStream_t stream) {
  (void)n_in; (void)out_size; (void)d_ws; (void)ws_size;
  const float* x = (const float*)d_in[0];
  const float* W = (const float*)d_in[1];
  const float* b = (const float*)d_in[2];
  float* out = (float*)d_out;

  int Bsz = in_sizes[0] / K_DIM;      // 262144
  int n_tiles = Bsz / TILE_M;         // 16384
  // Pick a grid so every wave runs the same trip count (safe __syncthreads).
  int grid = 512;
  while (grid > 1 && (n_tiles % (grid * WAVES_PER_BLOCK)) != 0) grid >>= 1;

  svdo_fused_kernel<<<dim3(grid), dim3(THREADS), 0, stream>>>(x, W, b, out,
                                                              n_tiles);
}